// LightGCN_30459908063509
// MI455X (gfx1250) — compile-verified
//
#include <hip/hip_runtime.h>

#define N_USERS 100000
#define N_ITEMS 50000
#define N_NODES (N_USERS + N_ITEMS)
#define DIM 64
#define N_EDGES 4800000
#define BATCH 131072

typedef __attribute__((ext_vector_type(2))) float v2f;
typedef __attribute__((ext_vector_type(8))) float v8f;

// ---------------------------------------------------------------------------
// init: buf0 = concat(user_emb, item_emb); acc = buf0; buf1 = 0
// float4-vectorized over the 38.4MB node table (L2-resident on MI455X).
// ---------------------------------------------------------------------------
__global__ void lgcn_init(const float* __restrict__ ue, const float* __restrict__ ie,
                          float* __restrict__ acc, float* __restrict__ buf0,
                          float* __restrict__ buf1) {
    size_t i = (size_t)blockIdx.x * blockDim.x + threadIdx.x;   // float4 index
    const size_t total = (size_t)N_NODES * DIM / 4;
    if (i >= total) return;
    const size_t uelems = (size_t)N_USERS * DIM / 4;
    float4 v = (i < uelems) ? ((const float4*)ue)[i]
                            : ((const float4*)ie)[i - uelems];
    ((float4*)buf0)[i] = v;
    ((float4*)acc)[i]  = v;
    ((float4*)buf1)[i] = make_float4(0.f, 0.f, 0.f, 0.f);
}

// ---------------------------------------------------------------------------
// SpMM scatter: y[dst] += w * x[src].  16 lanes per edge, float4 per lane.
// Edge arrays stream (57.6MB); node rows gather/scatter inside L2.
// ---------------------------------------------------------------------------
__global__ void lgcn_spmm(const int* __restrict__ src, const int* __restrict__ dst,
                          const float* __restrict__ w,
                          const float* __restrict__ x, float* __restrict__ y) {
    size_t tid = (size_t)blockIdx.x * blockDim.x + threadIdx.x;
    if (tid >= (size_t)N_EDGES * 16) return;
    int e    = (int)(tid >> 4);
    int part = (int)(tid & 15);
    int s = src[e];
    int d = dst[e];
    float we = w[e];
    float4 v = ((const float4*)(x + (size_t)s * DIM))[part];
    float* o = y + (size_t)d * DIM + (size_t)part * 4;
    atomicAdd(o + 0, we * v.x);
    atomicAdd(o + 1, we * v.y);
    atomicAdd(o + 2, we * v.z);
    atomicAdd(o + 3, we * v.w);
}

// ---------------------------------------------------------------------------
// fold: acc += layer; optionally zero the buffer that becomes next layer's out
// ---------------------------------------------------------------------------
__global__ void lgcn_fold(float* __restrict__ acc, const float* __restrict__ layer,
                          float* __restrict__ tozero, int do_zero) {
    size_t i = (size_t)blockIdx.x * blockDim.x + threadIdx.x;
    const size_t total = (size_t)N_NODES * DIM / 4;
    if (i >= total) return;
    float4 a = ((float4*)acc)[i];
    float4 l = ((const float4*)layer)[i];
    a.x += l.x; a.y += l.y; a.z += l.z; a.w += l.w;
    ((float4*)acc)[i] = a;
    if (do_zero) ((float4*)tozero)[i] = make_float4(0.f, 0.f, 0.f, 0.f);
}

// ---------------------------------------------------------------------------
// scoring: gamma[b] = (1/16) * dot(acc[u_b], acc[N_USERS+i_b]) over DIM=64.
// One wave32 per 16 batch rows: D(16x16) = U(16x64) x I^T(64x16) via a chain
// of 16 V_WMMA_F32_16X16X4_F32; emit the diagonal.
//
// Per ISA 7.12.2 (f32, wave32):
//   A 16x4 : VGPR0 = K { 0 (lanes<16) | 2 (lanes>=16) }, VGPR1 = K+1
//   B 4x16 : VGPR0 = K-row { 0 | 2 }, VGPR1 = K-row+1, col = lane%16
//   => per K-chunk both fragments are contiguous float2 loads.
//   C/D    : VGPR v, lane l -> D[v + 8*(l>=16)][l%16]
//   diag: lanes 0..7 -> c[l]; lanes 24..31 -> c[l-24]
// EXEC all-ones (BATCH % 16 == 0, full waves), as WMMA requires.
// ---------------------------------------------------------------------------
__global__ void lgcn_score(const float* __restrict__ acc, const int* __restrict__ users,
                           const int* __restrict__ items, float* __restrict__ out) {
    int wave = (int)(((size_t)blockIdx.x * blockDim.x + threadIdx.x) >> 5);
    int lane = threadIdx.x & 31;
    int base = wave * 16;                  // 16 batch rows per wave
    int n    = lane & 15;                  // A row m / B col n handled by lane
    int hi   = lane >> 4;                  // half-wave selects K phase

    int u = users[base + n];
    int it = items[base + n];
    const float* urow = acc + (size_t)u * DIM;                    // A[m][*]
    const float* irow = acc + (size_t)(N_USERS + it) * DIM;       // B[*][n]

    v8f c = {};
    #pragma unroll
    for (int kk = 0; kk < DIM; kk += 4) {
        int ka = kk + 2 * hi;
        v2f a; a.x = urow[ka]; a.y = urow[ka + 1];
        v2f b; b.x = irow[ka]; b.y = irow[ka + 1];
        c = __builtin_amdgcn_wmma_f32_16x16x4_f32(
                /*neg_a=*/false, a, /*neg_b=*/false, b,
                /*c_mod=*/(short)0, c, /*reuse_a=*/false, /*reuse_b=*/false);
    }

    // diagonal of D, scaled by (1/4)*(1/4) layer-mean normalization
    if (lane < 8)        out[base + lane]        = c[lane]      * 0.0625f;
    else if (lane >= 24) out[base + (lane - 16)] = c[lane - 24] * 0.0625f;
}

// ---------------------------------------------------------------------------
extern "C" void kernel_launch(void* const* d_in, const int* in_sizes, int n_in,
                              void* d_out, int out_size, void* d_ws, size_t ws_size,
                              hipStream_t stream) {
    const float* user_emb = (const float*)d_in[0];
    const float* item_emb = (const float*)d_in[1];
    const int*   edge_src = (const int*)d_in[2];
    const int*   edge_dst = (const int*)d_in[3];
    const float* edge_w   = (const float*)d_in[4];
    const int*   users    = (const int*)d_in[5];
    const int*   items    = (const int*)d_in[6];
    float*       out      = (float*)d_out;

    const size_t nodeBytes = (size_t)N_NODES * DIM * sizeof(float);   // 38.4 MB
    float* acc  = (float*)d_ws;
    float* buf0 = (float*)((char*)d_ws + nodeBytes);
    float* buf1 = (float*)((char*)d_ws + 2 * nodeBytes);

    dim3 blk(256);
    const unsigned elem4   = (unsigned)((size_t)N_NODES * DIM / 4);
    dim3 grdElem((elem4 + 255u) / 256u);
    dim3 grdEdge((unsigned)(((size_t)N_EDGES * 16 + 255) / 256));     // 300000 blocks

    lgcn_init<<<grdElem, blk, 0, stream>>>(user_emb, item_emb, acc, buf0, buf1);

    // layer 1: buf0 -> buf1
    lgcn_spmm<<<grdEdge, blk, 0, stream>>>(edge_src, edge_dst, edge_w, buf0, buf1);
    lgcn_fold<<<grdElem, blk, 0, stream>>>(acc, buf1, buf0, 1);
    // layer 2: buf1 -> buf0
    lgcn_spmm<<<grdEdge, blk, 0, stream>>>(edge_src, edge_dst, edge_w, buf1, buf0);
    lgcn_fold<<<grdElem, blk, 0, stream>>>(acc, buf0, buf1, 1);
    // layer 3: buf0 -> buf1
    lgcn_spmm<<<grdEdge, blk, 0, stream>>>(edge_src, edge_dst, edge_w, buf0, buf1);
    lgcn_fold<<<grdElem, blk, 0, stream>>>(acc, buf1, buf1, 0);

    // scoring: 8192 tiles of 16 rows; 8 waves/block -> 1024 blocks
    lgcn_score<<<dim3(BATCH / 16 / 8), blk, 0, stream>>>(acc, users, items, out);
}